// WaveCell_60979945668797
// MI455X (gfx1250) — compile-verified
//
#include <hip/hip_runtime.h>

typedef __attribute__((ext_vector_type(2))) float v2f;
typedef __attribute__((ext_vector_type(8))) float v8f;

#define WNX 512
#define WNY 512
#define WNT 300
#define NTI (WNX / 16)
#define NTJ (WNY / 16)

// PML damping ramp along one axis: b_vals[j] = 0.5*(j/20)^3
// rows/cols 0..20 get b_vals[20-i]; rows/cols n-22..n-2 get b_vals[i-(n-22)]; else 0.
__device__ __forceinline__ float pml_ramp(int i, int n) {
    float j = -1.0f;
    if (i <= 20)                        j = (float)(20 - i);
    else if (i >= n - 22 && i <= n - 2) j = (float)(i - (n - 22));
    if (j < 0.0f) return 0.0f;
    float t = j * 0.05f;           // j/20
    return 0.5f * t * t * t;       // PML_MAX * t^PML_P
}

// Tridiagonal Laplacian block D = tridiag(1,-2,1), 16x16 (symmetric)
__device__ __forceinline__ float dcoef(int r, int c) {
    int d = r - c;
    if (d == 0) return -2.0f;
    if (d == 1 || d == -1) return 1.0f;
    return 0.0f;
}

// B-operand fragment (rows k2,k2+1 of the 16x16 tile, column ln per lane)
__device__ __forceinline__ v2f load_bfrag(const float* __restrict__ u,
                                          int R0, int C0, int k2, int ln) {
    v2f b;
    b.x = u[(R0 + k2)     * WNY + C0 + ln];
    b.y = u[(R0 + k2 + 1) * WNY + C0 + ln];
    return b;
}

// A-operand fragment (row ln, consecutive cols k2,k2+1 -> one 8B load)
__device__ __forceinline__ v2f load_afrag(const float* __restrict__ u,
                                          int R0, int C0, int k2, int ln) {
    const float* p = &u[(R0 + ln) * WNY + C0 + k2];   // C0 mult of 16, k2 even -> 8B aligned
    return *(const v2f*)p;
}

__device__ __forceinline__ v8f wmma4(v2f a, v2f b, v8f c) {
    return __builtin_amdgcn_wmma_f32_16x16x4_f32(false, a, false, b, (short)0, c,
                                                 false, false);
}

// One-shot init: zero both ping-pong field buffers and precompute the
// time-invariant PML coefficients  C1 = 1/(4+b),  C3 = 4-b  (dt = 0.5).
__global__ __launch_bounds__(256)
void WaveCell_init(float* __restrict__ bufA, float* __restrict__ bufB,
                   float* __restrict__ c1buf, float* __restrict__ c3buf) {
    int i = blockIdx.x * 256 + threadIdx.x;
    if (i >= WNX * WNY) return;
    bufA[i] = 0.0f;
    bufB[i] = 0.0f;
    const int row = i / WNY, col = i - row * WNY;
    const float bx = pml_ramp(row, WNX);
    const float by = pml_ramp(col, WNY);
    const float b  = sqrtf(bx * bx + by * by);
    c1buf[i] = 1.0f / (4.0f + b);
    c3buf[i] = 4.0f - b;
}

// One leapfrog step:  out = C1*(8*un1 - C3*un2 + c2*lap(un1) + mask_src*x[t])
// Dense part of lap = D@U + U@D on the WMMA pipe (D symmetric tridiag block);
// the four rank-1 halo contributions are exact single-element adds in the epilogue.
__global__ __launch_bounds__(32)
void WaveCell_step(const float* __restrict__ un1, const float* __restrict__ un2,
                   float* __restrict__ out, const float* __restrict__ c2,
                   const float* __restrict__ msrc, const float* __restrict__ xsrc,
                   const float* __restrict__ c1buf, const float* __restrict__ c3buf,
                   int t)
{
    const int lane = threadIdx.x & 31;
    const int ln   = lane & 15;     // column index N (and row index M for A frags)
    const int hi   = lane >> 4;     // half-wave select
    const int TI = blockIdx.y, TJ = blockIdx.x;
    const int R0 = TI << 4, C0 = TJ << 4;

    v8f acc = {};

    // ---- dense stencil: D @ U_C  +  U_C @ D  (K=16 via 4 chained k-steps) ----
    // D is symmetric, so the same per-lane fragment serves as A operand (row ln,
    // cols k2,k2+1) and as B operand (rows k2,k2+1, col ln).
#pragma unroll
    for (int k = 0; k < 4; ++k) {
        const int k2 = 4 * k + 2 * hi;
        v2f d;  d.x = dcoef(ln, k2);  d.y = dcoef(ln, k2 + 1);
        v2f bU = load_bfrag(un1, R0, C0, k2, ln);
        acc = wmma4(d, bU, acc);                       // D @ U_C
        v2f aU = load_afrag(un1, R0, C0, k2, ln);
        acc = wmma4(aU, d, acc);                       // U_C @ D
    }

    // ---- elementwise PML update (C/D layout: VGPR r -> row r + 8*hi, col ln) ----
    const float xi = xsrc[t];
#pragma unroll
    for (int r = 0; r < 8; ++r) {
        const int row = R0 + r + 8 * hi;
        const int col = C0 + ln;
        const int idx = row * WNY + col;

        float lap = acc[r];
        // rank-1 halo terms (zero-padded boundary => skip at grid edges)
        if (TI > 0      && hi == 0 && r == 0) lap += un1[(R0 - 1)  * WNY + col];
        if (TI < NTI-1  && hi == 1 && r == 7) lap += un1[(R0 + 16) * WNY + col];
        if (TJ > 0      && ln == 0)           lap += un1[row * WNY + C0 - 1];
        if (TJ < NTJ-1  && ln == 15)          lap += un1[row * WNY + C0 + 16];
        // inv_h2 = 1 (H = 1)

        const float u1 = un1[idx];
        const float u2 = un2[idx];
        out[idx] = c1buf[idx] *
                   (8.0f * u1 - c3buf[idx] * u2 + c2[idx] * lap + msrc[idx] * xi);
    }
}

extern "C" void kernel_launch(void* const* d_in, const int* in_sizes, int n_in,
                              void* d_out, int out_size, void* d_ws, size_t ws_size,
                              hipStream_t stream) {
    const float* x    = (const float*)d_in[0];   // (NT,)
    const float* c2   = (const float*)d_in[1];   // (NX,NY)
    const float* msrc = (const float*)d_in[2];   // (NX,NY)
    // d_in[3] = mask_probe: unused (reference returns the final field, not a probe trace)

    float* bufA  = (float*)d_ws;
    float* bufB  = bufA + WNX * WNY;
    float* c1buf = bufB + WNX * WNY;
    float* c3buf = c1buf + WNX * WNY;
    float* bufs[2] = {bufA, bufB};

    const int n = WNX * WNY;
    WaveCell_init<<<(n + 255) / 256, 256, 0, stream>>>(bufA, bufB, c1buf, c3buf);

    dim3 grid(NTJ, NTI);
    for (int t = 0; t < WNT; ++t) {
        const float* un1 = bufs[t & 1];          // previous field
        float*       dst = bufs[1 - (t & 1)];    // holds un2; overwritten with un
        const float* un2 = dst;
        float*       out = (t == WNT - 1) ? (float*)d_out : dst;
        WaveCell_step<<<grid, 32, 0, stream>>>(un1, un2, out, c2, msrc, x,
                                               c1buf, c3buf, t);
    }
}